// BERT_ner_28295244546140
// MI455X (gfx1250) — compile-verified
//
#include <hip/hip_runtime.h>
#include <hip/hip_bf16.h>

typedef float v2f __attribute__((ext_vector_type(2)));
typedef float v8f __attribute__((ext_vector_type(8)));

#define S_LEN 512
#define B_SZ  4096
#define T_SZ  16

#define LOG2E 1.4426950408889634f
#define LN2   0.6931471805599453f

// ---------------------------------------------------------------------------
// Kernel 1: gold-path score (gathers). One thread per batch element.
// ---------------------------------------------------------------------------
__global__ void crf_gold_score(const float* __restrict__ em,
                               const int*   __restrict__ tags,
                               const float* __restrict__ masks,
                               const float* __restrict__ start_tr,
                               const float* __restrict__ end_tr,
                               const float* __restrict__ trans,
                               float*       __restrict__ gold) {
  int b = blockIdx.x * blockDim.x + threadIdx.x;
  if (b >= B_SZ) return;

  int prev = tags[b];
  float sc = start_tr[prev] + em[(size_t)b * T_SZ + prev];
  float msum = masks[b];

  for (int t = 1; t < S_LEN; ++t) {
    int   tag = tags[t * B_SZ + b];
    float mk  = masks[t * B_SZ + b];
    sc += (trans[prev * T_SZ + tag] + em[((size_t)t * B_SZ + b) * T_SZ + tag]) * mk;
    msum += mk;
    prev = tag;
  }
  int send = (int)msum - 1;                  // index of last valid step
  int last = tags[send * B_SZ + b];
  gold[b] = sc + end_tr[last];
}

// ---------------------------------------------------------------------------
// Kernel 2: forward-algorithm normalizer via f32 WMMA, base-2 log domain.
//
// Per wave: a tile of 16 batch elements. State vector layout == WMMA C/D
// layout of D^T = E^T x P^T (16x16 f32 via 4 chained v_wmma_f32_16x16x4_f32):
//   lane L in [0,16):  sv[r] = log2e*score[b=tile*16+L][state j=r],   r=0..7
//   lane L in [16,32): sv[r] = log2e*score[b=tile*16+L-16][state j=r+8]
// E = exp(transitions) lives in A-operand registers for the whole scan.
// Raw v_exp_f32 / v_log_f32 (base-2, no libm fixups) are safe here: the log
// argument is sum_i p_i*E_ij with max p = 1, E in [0.9,1.1] -> in [0.9,17.6].
// ---------------------------------------------------------------------------
__global__ void __launch_bounds__(256)
crf_forward_wmma(const float* __restrict__ em,
                 const float* __restrict__ masks,
                 const float* __restrict__ start_tr,
                 const float* __restrict__ end_tr,
                 const float* __restrict__ trans,
                 const float* __restrict__ gold,
                 float*       __restrict__ out) {
  const int lane = threadIdx.x & 31;
  const int wave = threadIdx.x >> 5;
  const int tile = blockIdx.x * (blockDim.x >> 5) + wave;   // 16-batch tile
  const int hi   = lane >> 4;                               // half-wave select
  const int col  = lane & 15;
  const int b    = tile * 16 + col;                         // my batch element
  const int j0   = hi * 8;                                  // my state offset

  // --- A-operand: E^T in 16x4 K-chunks.  A chunk c, VGPR0 holds K=4c (lanes
  // 0-15) / K=4c+2 (lanes 16-31); VGPR1 holds K=4c+1 / 4c+3.  M = col = j.
  v2f a[4];
#pragma unroll
  for (int c = 0; c < 4; ++c) {
    int i0 = 4 * c + 2 * hi;
    a[c].x = __expf(trans[i0 * T_SZ + col]);
    a[c].y = __expf(trans[(i0 + 1) * T_SZ + col]);
  }

  // --- init: score0 = (start + emissions[0]) in base-2 units
  float sv[8];
#pragma unroll
  for (int r = 0; r < 8; ++r)
    sv[r] = (start_tr[j0 + r] + em[((size_t)b) * T_SZ + j0 + r]) * LOG2E;

  float e_end[8];
#pragma unroll
  for (int r = 0; r < 8; ++r) e_end[r] = end_tr[j0 + r];

  for (int t = 1; t < S_LEN; ++t) {
    // m_b = max over this batch's 16 states (8 local + cross-half exchange)
    float m = sv[0];
#pragma unroll
    for (int r = 1; r < 8; ++r) m = fmaxf(m, sv[r]);
    m = fmaxf(m, __shfl_xor(m, 16, 32));

    // p = 2^(sv - m)  == exp(score - max)   (raw v_exp_f32)
    float p[8];
#pragma unroll
    for (int r = 0; r < 8; ++r) p[r] = __builtin_amdgcn_exp2f(sv[r] - m);

    // 4 cross-half exchanges serve all 8 opposite-half B-operand slots:
    // pre-select so each shuffle carries what the *other* half needs.
    float s0 = __shfl_xor(hi ? p[0] : p[2], 16, 32);
    float s1 = __shfl_xor(hi ? p[1] : p[3], 16, 32);
    float s2 = __shfl_xor(hi ? p[4] : p[6], 16, 32);
    float s3 = __shfl_xor(hi ? p[5] : p[7], 16, 32);

    // B chunk c, VGPR0: K=4c (lanes 0-15) / K=4c+2 (lanes 16-31), N = batch
    v2f bm[4];
    bm[0].x = hi ? s0 : p[0];  bm[0].y = hi ? s1 : p[1];   // K=0..3
    bm[1].x = hi ? s2 : p[4];  bm[1].y = hi ? s3 : p[5];   // K=4..7
    bm[2].x = hi ? p[2] : s0;  bm[2].y = hi ? p[3] : s1;   // K=8..11
    bm[3].x = hi ? p[6] : s2;  bm[3].y = hi ? p[7] : s3;   // K=12..15

    // D^T = E^T x P^T : 4 chained f32 WMMAs over K
    v8f acc = {};
#pragma unroll
    for (int c = 0; c < 4; ++c)
      acc = __builtin_amdgcn_wmma_f32_16x16x4_f32(
          /*neg_a=*/false, a[c], /*neg_b=*/false, bm[c],
          /*c_mod=*/(short)0, acc, /*reuse_a=*/false, /*reuse_b=*/false);

    // emissions tile: each 16-batch tile reads a contiguous 1KB block
    const float* ep = em + ((size_t)t * B_SZ + b) * T_SZ + j0;
    __builtin_prefetch(ep + (size_t)B_SZ * T_SZ, 0, 1);   // next step's tile
    float4 e0 = *(const float4*)(ep);
    float4 e1 = *(const float4*)(ep + 4);
    float  mk = masks[t * B_SZ + b];
    float  em8[8] = {e0.x, e0.y, e0.z, e0.w, e1.x, e1.y, e1.z, e1.w};

    if (mk > 0.f) {                       // jnp.where(mask, next, prev)
#pragma unroll
      for (int r = 0; r < 8; ++r)
        sv[r] = fmaf(em8[r], LOG2E, m + __builtin_amdgcn_logf(acc[r]));
    }
  }

  // final: logsumexp(sv + end_transitions) over the 16 states, back to ln
#pragma unroll
  for (int r = 0; r < 8; ++r) sv[r] = fmaf(e_end[r], LOG2E, sv[r]);
  float m = sv[0];
#pragma unroll
  for (int r = 1; r < 8; ++r) m = fmaxf(m, sv[r]);
  m = fmaxf(m, __shfl_xor(m, 16, 32));
  float s = 0.f;
#pragma unroll
  for (int r = 0; r < 8; ++r) s += __builtin_amdgcn_exp2f(sv[r] - m);
  s += __shfl_xor(s, 16, 32);
  float norm = (m + __builtin_amdgcn_logf(s)) * LN2;

  if (!hi) out[b] = gold[b] - norm;       // one writer per batch element
}

// ---------------------------------------------------------------------------
extern "C" void kernel_launch(void* const* d_in, const int* in_sizes, int n_in,
                              void* d_out, int out_size, void* d_ws, size_t ws_size,
                              hipStream_t stream) {
  const float* em       = (const float*)d_in[0];
  const int*   tags     = (const int*)d_in[1];
  const float* masks    = (const float*)d_in[2];
  const float* start_tr = (const float*)d_in[3];
  const float* end_tr   = (const float*)d_in[4];
  const float* trans    = (const float*)d_in[5];
  float* out  = (float*)d_out;
  float* gold = (float*)d_ws;             // B_SZ floats of scratch

  crf_gold_score<<<B_SZ / 256, 256, 0, stream>>>(
      em, tags, masks, start_tr, end_tr, trans, gold);

  // 256 tiles of 16 batches; 8 wave32 per 256-thread block -> 32 blocks
  crf_forward_wmma<<<(B_SZ / 16) / 8, 256, 0, stream>>>(
      em, masks, start_tr, end_tr, trans, gold, out);
}